// FM_12060268167845
// MI455X (gfx1250) — compile-verified
//
#include <hip/hip_runtime.h>
#include <stdint.h>

// FM forward: memory-bound gather kernel for MI455X (gfx1250).
// V-row gathers staged LDS-side via CDNA5 async-to-LDS loads (ASYNCcnt),
// double-buffered, one 512B V row per global_load_async_to_lds_b128 (GVS
// saddr form: 64-bit base in SGPRs + 32-bit per-lane offset).

#define B_ROWS    8192
#define K_ACT     200
#define FDIM      128
#define WAVES     8            // waves per block (wave32)
#define CH        8            // V rows per async chunk (8 * 512B = 4KB)
#define NCHUNK    (K_ACT / CH) // 25, exact

typedef float float4v __attribute__((ext_vector_type(4)));
typedef int   int4v   __attribute__((ext_vector_type(4)));

// Workgroup-relative LDS byte offset of a __shared__ object.
__device__ __forceinline__ uint32_t lds_off(const void* p) {
    return (uint32_t)(uintptr_t)p;
}

__device__ __forceinline__ float wave_sum(float v) {
    #pragma unroll
    for (int m = 16; m; m >>= 1) v += __shfl_xor(v, m, 32);
    return v;
}

// Async gather of one 512B V row: 32 lanes x b128. GVS addressing:
// mem = SADDR(V base) + sext(voff32); LDS dst = lds32 (lane*16 folded in).
__device__ __forceinline__ void async_gather_row(uint32_t lds32, uint32_t voff32,
                                                 const float* vbase) {
    asm volatile("global_load_async_to_lds_b128 %0, %1, %2"
                 :: "v"(lds32), "v"(voff32), "s"(vbase) : "memory");
}

__global__ __launch_bounds__(WAVES * 32, 1)
void fm_fwd_kernel(const int* __restrict__ idx,
                   const float* __restrict__ x_vals,
                   const float* __restrict__ b_vals,
                   const float* __restrict__ w,
                   const float* __restrict__ V,
                   const float* __restrict__ bias,
                   float* __restrict__ out)
{
    __shared__ __attribute__((aligned(16))) float ldsV[WAVES][2][CH * FDIM]; // 64 KB
    __shared__ __attribute__((aligned(16))) int   ldsI[WAVES][K_ACT];
    __shared__ __attribute__((aligned(16))) float ldsX[WAVES][K_ACT];

    const int lane = threadIdx.x & 31;
    const int wv   = threadIdx.x >> 5;

    // Row index is wave-uniform: pin it in an SGPR so all row-base pointer
    // arithmetic stays scalar AND in the global address space ->
    // global_load_b32 v, vLane, s[base] scale_offset (saddr form, no flat).
    const int row = __builtin_amdgcn_readfirstlane((int)(blockIdx.x * WAVES + wv));

    const int*   idxr = idx    + (size_t)row * K_ACT;
    const float* xr   = x_vals + (size_t)row * K_ACT;
    const float* br   = b_vals + (size_t)row * K_ACT;

    // ---- Stage idx/x rows into LDS (coalesced) + scalar reductions ----
    float sx = 0.0f, sbw = 0.0f;
    #pragma unroll
    for (int i = 0; i < 7; ++i) {                 // ceil(200/32) uniform trips
        const int k = lane + i * 32;
        if (k < K_ACT) {
            int   iv = idxr[k];
            float xk = xr[k];
            ldsI[wv][k] = iv;
            ldsX[wv][k] = xk;
            sx  += xk;
            sbw += br[k] * w[iv];                 // w is 4MB -> L2-resident
        }
    }
    sx  = wave_sum(sx);
    sbw = wave_sum(sbw);

    const uint32_t lbase0 = lds_off(&ldsV[wv][0][0]);
    const uint32_t lbase1 = lds_off(&ldsV[wv][1][0]);
    const uint32_t laneb  = (uint32_t)lane * 16u;   // 16B/lane -> 512B per row

    // ---- Prologue: async-gather chunk 0 into LDS buffer 0 ----
    {
        const int4v i0 = *(const int4v*)&ldsI[wv][0];   // uniform b128 broadcast
        const int4v i1 = *(const int4v*)&ldsI[wv][4];
        const uint32_t lb = lbase0 + laneb;
        async_gather_row(lb + 0 * 512u, ((uint32_t)i0.x << 9) + laneb, V);
        async_gather_row(lb + 1 * 512u, ((uint32_t)i0.y << 9) + laneb, V);
        async_gather_row(lb + 2 * 512u, ((uint32_t)i0.z << 9) + laneb, V);
        async_gather_row(lb + 3 * 512u, ((uint32_t)i0.w << 9) + laneb, V);
        async_gather_row(lb + 4 * 512u, ((uint32_t)i1.x << 9) + laneb, V);
        async_gather_row(lb + 5 * 512u, ((uint32_t)i1.y << 9) + laneb, V);
        async_gather_row(lb + 6 * 512u, ((uint32_t)i1.z << 9) + laneb, V);
        async_gather_row(lb + 7 * 512u, ((uint32_t)i1.w << 9) + laneb, V);
    }

    float4v a1 = {0.f, 0.f, 0.f, 0.f};   // XV   chunk (4 f per lane)
    float4v a2 = {0.f, 0.f, 0.f, 0.f};   // X2V2 chunk

    for (int c = 0; c < NCHUNK; ++c) {
        // Issue next chunk into the other buffer, then wait for current chunk.
        if (c + 1 < NCHUNK) {
            const uint32_t nb = (((c + 1) & 1) ? lbase1 : lbase0) + laneb;
            const int kb = (c + 1) * CH;
            const int4v i0 = *(const int4v*)&ldsI[wv][kb];
            const int4v i1 = *(const int4v*)&ldsI[wv][kb + 4];
            async_gather_row(nb + 0 * 512u, ((uint32_t)i0.x << 9) + laneb, V);
            async_gather_row(nb + 1 * 512u, ((uint32_t)i0.y << 9) + laneb, V);
            async_gather_row(nb + 2 * 512u, ((uint32_t)i0.z << 9) + laneb, V);
            async_gather_row(nb + 3 * 512u, ((uint32_t)i0.w << 9) + laneb, V);
            async_gather_row(nb + 4 * 512u, ((uint32_t)i1.x << 9) + laneb, V);
            async_gather_row(nb + 5 * 512u, ((uint32_t)i1.y << 9) + laneb, V);
            async_gather_row(nb + 6 * 512u, ((uint32_t)i1.z << 9) + laneb, V);
            async_gather_row(nb + 7 * 512u, ((uint32_t)i1.w << 9) + laneb, V);
            // 16 outstanding; async loads complete in order -> <=8 means chunk c landed.
            asm volatile("s_wait_asynccnt 8" ::: "memory");
        } else {
            asm volatile("s_wait_asynccnt 0" ::: "memory");
        }

        const int kb = c * CH;
        const float* buf = (c & 1) ? &ldsV[wv][1][0] : &ldsV[wv][0][0];
        const float4v x0 = *(const float4v*)&ldsX[wv][kb];      // uniform b128
        const float4v x1 = *(const float4v*)&ldsX[wv][kb + 4];  // broadcasts
        const float xs[CH] = {x0.x, x0.y, x0.z, x0.w, x1.x, x1.y, x1.z, x1.w};
        #pragma unroll
        for (int kk = 0; kk < CH; ++kk) {
            float x = xs[kk];
            float4v v = *(const float4v*)(buf + kk * FDIM + lane * 4); // ds_load_b128
            a1 += x * v;
            a2 += (x * x) * (v * v);
        }
    }

    // ---- p = 0.5 / sx * sum_f (XV^2 - X2V2) ----
    float part = (a1.x * a1.x - a2.x) + (a1.y * a1.y - a2.y)
               + (a1.z * a1.z - a2.z) + (a1.w * a1.w - a2.w);
    part = wave_sum(part);

    if (lane == 0) {
        float logit = bias[0] + sbw + 0.5f * (1.0f / sx) * part;
        out[row] = 1.0f / (1.0f + __expf(-logit));
    }
}

extern "C" void kernel_launch(void* const* d_in, const int* in_sizes, int n_in,
                              void* d_out, int out_size, void* d_ws, size_t ws_size,
                              hipStream_t stream) {
    const int*   idx    = (const int*)d_in[0];
    const float* x_vals = (const float*)d_in[1];
    const float* b_vals = (const float*)d_in[2];
    const float* w      = (const float*)d_in[3];
    const float* V      = (const float*)d_in[4];
    const float* bias   = (const float*)d_in[5];
    float* out = (float*)d_out;

    dim3 grid(B_ROWS / WAVES);   // 1024 blocks
    dim3 block(WAVES * 32);      // 256 threads = 8 wave32
    hipLaunchKernelGGL(fm_fwd_kernel, grid, block, 0, stream,
                       idx, x_vals, b_vals, w, V, bias, out);
}